// sparseLayer_7258494730807
// MI455X (gfx1250) — compile-verified
//
#include <hip/hip_runtime.h>

// out[b,o] = sum_{r<4} x[b, 4o+r]*w[4o+r] + sum_r bias[4o+r]
//
// HBM-streaming kernel: 335 MB moved, 0.134 GFLOP -> roofline ~14.4 us
// @ 23.3 TB/s. Fast path: 32-bit indices (saddr+voffset addressing), 4
// non-temporal b128 loads in flight per lane, w/bias loaded ONCE per thread
// (all 4 groups share o because the stream span is a multiple of n_out),
// no bounds checks when the launch divides evenly, NT stores.

typedef float v4f __attribute__((ext_vector_type(4)));

// Fast path, no bounds checks. Preconditions (host-enforced):
//   total % (256*4) == 0, quarter = total/4, quarter % n_out == 0,
//   n_out power of two, all byte offsets < 2^31.
__global__ __launch_bounds__(256) void grouped_wsum4_exact_kernel(
    const v4f* __restrict__ x4,     // [total] groups of 4 floats
    const v4f* __restrict__ w4,     // [n_out]
    const v4f* __restrict__ b4,     // [n_out]
    float* __restrict__ out,        // [total]
    unsigned o_mask,                // n_out - 1
    unsigned quarter)               // total / 4 (multiple of n_out)
{
    const unsigned tid = blockIdx.x * 256u + threadIdx.x;
    // quarter % n_out == 0  =>  (tid + k*quarter) & o_mask == tid & o_mask.
    const unsigned o = tid & o_mask;

    // Issue all 4 streaming loads before consuming any (independent b128 NT).
    v4f xv[4];
#pragma unroll
    for (int k = 0; k < 4; ++k)
        xv[k] = __builtin_nontemporal_load(&x4[tid + (unsigned)k * quarter]);

    const v4f w = w4[o];            // cached: 16 KB table, hot in L2/WGP$
    const v4f b = b4[o];
    const float bsum = (b.x + b.y) + (b.z + b.w);

#pragma unroll
    for (int k = 0; k < 4; ++k) {
        float acc = bsum;
        acc = fmaf(xv[k].x, w.x, acc);
        acc = fmaf(xv[k].y, w.y, acc);
        acc = fmaf(xv[k].z, w.z, acc);
        acc = fmaf(xv[k].w, w.w, acc);
        __builtin_nontemporal_store(acc, &out[tid + (unsigned)k * quarter]);
    }
}

// Checked variant: ratio==4, n_out pow2, 32-bit offsets, any total.
__global__ __launch_bounds__(256) void grouped_wsum4_checked_kernel(
    const v4f* __restrict__ x4,
    const v4f* __restrict__ w4,
    const v4f* __restrict__ b4,
    float* __restrict__ out,
    unsigned o_mask,
    unsigned quarter,               // ceil(total/4)
    unsigned total)
{
    const unsigned tid = blockIdx.x * 256u + threadIdx.x;
    unsigned idx[4];
    v4f xv[4];
#pragma unroll
    for (int k = 0; k < 4; ++k) {
        idx[k] = tid + (unsigned)k * quarter;
        if (idx[k] < total)
            xv[k] = __builtin_nontemporal_load(&x4[idx[k]]);
    }
#pragma unroll
    for (int k = 0; k < 4; ++k) {
        if (idx[k] < total) {
            const unsigned o = idx[k] & o_mask;
            const v4f w = w4[o];
            const v4f b = b4[o];
            float acc = (b.x + b.y) + (b.z + b.w);
            acc = fmaf(xv[k].x, w.x, acc);
            acc = fmaf(xv[k].y, w.y, acc);
            acc = fmaf(xv[k].z, w.z, acc);
            acc = fmaf(xv[k].w, w.w, acc);
            __builtin_nontemporal_store(acc, &out[idx[k]]);
        }
    }
}

// Generic fallback: any ratio / n_out / size (64-bit safe).
__global__ __launch_bounds__(256) void grouped_wsum_generic_kernel(
    const float* __restrict__ x,
    const float* __restrict__ w,
    const float* __restrict__ bias,
    float* __restrict__ out,
    int n_out, int ratio, long long total)
{
    const long long stride = (long long)gridDim.x * blockDim.x;
    for (long long idx = (long long)blockIdx.x * blockDim.x + threadIdx.x;
         idx < total; idx += stride) {
        const int o = (int)(idx % n_out);
        const long long xbase = idx * ratio;
        const long long wbase = (long long)o * ratio;
        float acc = 0.0f;
        for (int r = 0; r < ratio; ++r) acc += bias[wbase + r];
        for (int r = 0; r < ratio; ++r)
            acc = fmaf(x[xbase + r], w[wbase + r], acc);
        __builtin_nontemporal_store(acc, &out[idx]);
    }
}

extern "C" void kernel_launch(void* const* d_in, const int* in_sizes, int n_in_args,
                              void* d_out, int out_size, void* d_ws, size_t ws_size,
                              hipStream_t stream) {
    (void)d_ws; (void)ws_size; (void)n_in_args;

    const float* x    = (const float*)d_in[0];
    const float* w    = (const float*)d_in[1];
    const float* bias = (const float*)d_in[2];
    float* out = (float*)d_out;

    // Derive shapes from sizes (n_out scalar lives in device memory; cannot be
    // read synchronously under graph capture).
    const long long x_elems = (long long)in_sizes[0];   // B * n_in
    const int n_in = in_sizes[1];                       // 4096
    const long long B = x_elems / n_in;                 // 16384
    const int n_out = (int)((long long)out_size / B);   // 1024
    const int ratio = n_in / n_out;                     // 4
    const long long total = (long long)out_size;        // 16,777,216

    const int threads = 256;
    const bool pow2 = (n_out & (n_out - 1)) == 0;
    const bool fits32 = (total * 16LL) < (1LL << 31);   // byte offsets < 2^31

    if (ratio == 4 && pow2 && fits32) {
        const unsigned utotal = (unsigned)total;
        const bool exact = (total % (threads * 4) == 0) &&
                           ((total / 4) % n_out == 0);
        if (exact) {
            const unsigned quarter = utotal / 4u;
            const unsigned blocks = quarter / (unsigned)threads;
            grouped_wsum4_exact_kernel<<<blocks, threads, 0, stream>>>(
                (const v4f*)x, (const v4f*)w, (const v4f*)bias, out,
                (unsigned)(n_out - 1), quarter);
        } else {
            const unsigned quarter = (utotal + 3u) / 4u;
            const unsigned blocks = (quarter + threads - 1) / threads;
            grouped_wsum4_checked_kernel<<<blocks, threads, 0, stream>>>(
                (const v4f*)x, (const v4f*)w, (const v4f*)bias, out,
                (unsigned)(n_out - 1), quarter, utotal);
        }
    } else {
        long long blocks = (total + threads - 1) / threads;
        if (blocks > 65535LL * 16) blocks = 65535LL * 16;
        grouped_wsum_generic_kernel<<<(int)blocks, threads, 0, stream>>>(
            x, w, bias, out, n_out, ratio, total);
    }
}